// GraphSAGE_50663434224272
// MI455X (gfx1250) — compile-verified
//
#include <hip/hip_runtime.h>
#include <hip/hip_bf16.h>

typedef __attribute__((ext_vector_type(2))) float v2f;
typedef __attribute__((ext_vector_type(8))) float v8f;

#define FD   64   // feature dim
#define LDW  66   // padded LDS row stride for transposed weights

// ---------------------------------------------------------------- CSR build
__global__ void sage_zero_i32(int* __restrict__ p, long n) {
    long i = (long)blockIdx.x * blockDim.x + threadIdx.x;
    if (i < n) p[i] = 0;
}

__global__ void sage_count_deg(const int* __restrict__ dst, int* __restrict__ deg, long E) {
    long i = (long)blockIdx.x * blockDim.x + threadIdx.x;
    if (i < E) atomicAdd(&deg[dst[i]], 1);
}

// per-256-block exclusive scan (Hillis-Steele), emit block totals
__global__ void sage_scan1(const int* __restrict__ in, int* __restrict__ excl,
                           int* __restrict__ blockSums, long n) {
    __shared__ int sh[256];
    const int  t = threadIdx.x;
    const long i = (long)blockIdx.x * 256 + t;
    const int  v = (i < n) ? in[i] : 0;
    sh[t] = v;
    __syncthreads();
    #pragma unroll
    for (int off = 1; off < 256; off <<= 1) {
        int u = (t >= off) ? sh[t - off] : 0;
        __syncthreads();
        sh[t] += u;
        __syncthreads();
    }
    if (i < n) excl[i] = sh[t] - v;
    if (t == 255) blockSums[blockIdx.x] = sh[255];
}

// single block scans the (<=256) block sums exclusively
__global__ void sage_scan2(int* __restrict__ blockSums, int nblocks) {
    __shared__ int sh[256];
    const int t = threadIdx.x;
    const int v = (t < nblocks) ? blockSums[t] : 0;
    sh[t] = v;
    __syncthreads();
    #pragma unroll
    for (int off = 1; off < 256; off <<= 1) {
        int u = (t >= off) ? sh[t - off] : 0;
        __syncthreads();
        sh[t] += u;
        __syncthreads();
    }
    if (t < nblocks) blockSums[t] = sh[t] - v;   // exclusive
}

// add block offsets; also initialize the placement cursors
__global__ void sage_scan3(int* __restrict__ excl, const int* __restrict__ blockSums,
                           int* __restrict__ cursor, long n) {
    const long i = (long)blockIdx.x * 256 + threadIdx.x;
    if (i < n) {
        const int r = excl[i] + blockSums[blockIdx.x];
        excl[i]   = r;
        cursor[i] = r;
    }
}

__global__ void sage_place(const int* __restrict__ src, const int* __restrict__ dst,
                           int* __restrict__ cursor, int* __restrict__ csr_src, long E) {
    long e = (long)blockIdx.x * blockDim.x + threadIdx.x;
    if (e < E) {
        const int pos = atomicAdd(&cursor[dst[e]], 1);
        csr_src[pos] = src[e];
    }
}

__global__ void sage_invdeg(const int* __restrict__ deg, float* __restrict__ invd, long n) {
    long i = (long)blockIdx.x * blockDim.x + threadIdx.x;
    if (i < n) invd[i] = 1.0f / fmaxf((float)deg[i], 1.0f);
}

// ------------------------------------------------- per-layer mean aggregation
// One wave per destination node; lane owns a float2 feature slice.
// Each neighbor row is a fully coalesced 256B wave read; no atomics.
__global__ __launch_bounds__(256) void sage_aggregate(
    const float* __restrict__ x, const int* __restrict__ rowstart,
    const int* __restrict__ deg, const int* __restrict__ csr_src,
    const float* __restrict__ invd, float* __restrict__ agg, long N) {
    const int  wave = threadIdx.x >> 5;
    const int  lane = threadIdx.x & 31;
    const long node = (long)blockIdx.x * 8 + wave;
    if (node >= N) return;
    const int beg = rowstart[node];
    const int cnt = deg[node];
    float2 acc = make_float2(0.0f, 0.0f);
    for (int j = 0; j < cnt; ++j) {
        const int s = csr_src[beg + j];                       // wave-uniform
        const float2 v = *(const float2*)(x + (long)s * FD + lane * 2);
        acc.x += v.x;
        acc.y += v.y;
    }
    const float sc = invd[node];
    acc.x *= sc;
    acc.y *= sc;
    *(float2*)(agg + node * FD + lane * 2) = acc;
}

// ----------------------------------------------------------- fused layer GEMM
// out = [relu]( agg @ Wl + x @ Wr + b ), fp32 WMMA 16x16x4.
// Weights staged transposed in LDS (sW[col*LDW + k]) so each B fragment is a
// contiguous 8B-aligned pair -> single ds_load_b64, no register shuffling.
__global__ __launch_bounds__(256) void sage_gemm(
    const float* __restrict__ agg, const float* __restrict__ xin,
    const float* __restrict__ Wl, const float* __restrict__ Wr,
    const float* __restrict__ bias, float* __restrict__ out,
    int do_relu) {
    __shared__ float sWl[FD * LDW];
    __shared__ float sWr[FD * LDW];
    const int tid = threadIdx.x;
    #pragma unroll 4
    for (int i = tid; i < FD * FD; i += 256) {
        const int k = i >> 6;       // row of W
        const int n = i & 63;       // col of W
        sWl[n * LDW + k] = Wl[i];
        sWr[n * LDW + k] = Wr[i];
    }
    __syncthreads();

    const int  wave = tid >> 5;
    const int  lane = tid & 31;
    const int  half = lane >> 4;   // selects K pair of the 16x4 fragment
    const int  l16  = lane & 15;
    const long rowbase = (long)blockIdx.x * 128 + (long)wave * 16;

    v8f acc[4];
    #pragma unroll
    for (int nt = 0; nt < 4; ++nt) {
        const float bv = bias[nt * 16 + l16];
        #pragma unroll
        for (int i = 0; i < 8; ++i) acc[nt][i] = bv;
    }

    const long arow = rowbase + l16;
    const float* aR = agg + arow * FD;
    const float* xR = xin + arow * FD;
    const int koff  = half * 2;

    // GEMM 1: agg @ Wl
    #pragma unroll
    for (int k = 0; k < FD; k += 4) {
        const v2f a = *(const v2f*)(aR + k + koff);
        #pragma unroll
        for (int nt = 0; nt < 4; ++nt) {
            const v2f b = *(const v2f*)(&sWl[(nt * 16 + l16) * LDW + k + koff]);
            acc[nt] = __builtin_amdgcn_wmma_f32_16x16x4_f32(
                false, a, false, b, (short)0, acc[nt], false, false);
        }
    }
    // GEMM 2: x @ Wr
    #pragma unroll
    for (int k = 0; k < FD; k += 4) {
        const v2f a = *(const v2f*)(xR + k + koff);
        #pragma unroll
        for (int nt = 0; nt < 4; ++nt) {
            const v2f b = *(const v2f*)(&sWr[(nt * 16 + l16) * LDW + k + koff]);
            acc[nt] = __builtin_amdgcn_wmma_f32_16x16x4_f32(
                false, a, false, b, (short)0, acc[nt], false, false);
        }
    }

    #pragma unroll
    for (int nt = 0; nt < 4; ++nt) {
        #pragma unroll
        for (int i = 0; i < 8; ++i) {
            const int m = i + half * 8;
            float v = acc[nt][i];
            if (do_relu) v = fmaxf(v, 0.0f);
            out[(rowbase + m) * FD + nt * 16 + l16] = v;
        }
    }
}

extern "C" void kernel_launch(void* const* d_in, const int* in_sizes, int n_in,
                              void* d_out, int out_size, void* d_ws, size_t ws_size,
                              hipStream_t stream) {
    const float* x   = (const float*)d_in[0];
    const int*   ei  = (const int*)d_in[1];
    const float* wl1 = (const float*)d_in[2];
    const float* b1  = (const float*)d_in[3];
    const float* wr1 = (const float*)d_in[4];
    const float* wl2 = (const float*)d_in[5];
    const float* b2  = (const float*)d_in[6];
    const float* wr2 = (const float*)d_in[7];
    const float* wl3 = (const float*)d_in[8];
    const float* b3  = (const float*)d_in[9];
    const float* wr3 = (const float*)d_in[10];

    const long N = (long)in_sizes[0] / FD;   // 65536
    const long E = (long)in_sizes[1] / 2;    // 1048576
    const int* src = ei;
    const int* dst = ei + E;

    // workspace layout
    char* w = (char*)d_ws;
    int*   deg       = (int*)w;            w += N * sizeof(int);
    int*   rowstart  = (int*)w;            w += N * sizeof(int);
    int*   cursor    = (int*)w;            w += N * sizeof(int);
    int*   blockSums = (int*)w;            w += 256 * sizeof(int);
    int*   csr_src   = (int*)w;            w += E * sizeof(int);
    float* invd      = (float*)w;          w += N * sizeof(float);
    float* agg       = (float*)w;          w += N * FD * sizeof(float);
    float* h1        = (float*)w;          w += N * FD * sizeof(float);
    float* h2        = (float*)w;
    float* out       = (float*)d_out;

    const int  B      = 256;
    const long gN     = (N + B - 1) / B;     // 256
    const long gE     = (E + B - 1) / B;
    const long gAgg   = (N + 7) / 8;         // one wave per node, 8 waves/block
    const long gGemm  = N / 128;             // 128 rows per block

    // ---- CSR build (once per launch) ----
    sage_zero_i32<<<gN, B, 0, stream>>>(deg, N);
    sage_count_deg<<<gE, B, 0, stream>>>(dst, deg, E);
    sage_scan1<<<gN, B, 0, stream>>>(deg, rowstart, blockSums, N);
    sage_scan2<<<1, B, 0, stream>>>(blockSums, (int)gN);
    sage_scan3<<<gN, B, 0, stream>>>(rowstart, blockSums, cursor, N);
    sage_invdeg<<<gN, B, 0, stream>>>(deg, invd, N);
    sage_place<<<gE, B, 0, stream>>>(src, dst, cursor, csr_src, E);

    // ---- layer 1 ----
    sage_aggregate<<<gAgg, B, 0, stream>>>(x, rowstart, deg, csr_src, invd, agg, N);
    sage_gemm<<<gGemm, B, 0, stream>>>(agg, x, wl1, wr1, b1, h1, 1);

    // ---- layer 2 ----
    sage_aggregate<<<gAgg, B, 0, stream>>>(h1, rowstart, deg, csr_src, invd, agg, N);
    sage_gemm<<<gGemm, B, 0, stream>>>(agg, h1, wl2, wr2, b2, h2, 1);

    // ---- layer 3 (no relu) -> d_out ----
    sage_aggregate<<<gAgg, B, 0, stream>>>(h2, rowstart, deg, csr_src, invd, agg, N);
    sage_gemm<<<gGemm, B, 0, stream>>>(agg, h2, wl3, wr3, b3, out, 0);
}